// Attention_35862976921794
// MI455X (gfx1250) — compile-verified
//
#include <hip/hip_runtime.h>
#include <hip/hip_bf16.h>
#include <stddef.h>

// Problem dims (fixed by the reference)
#define B_   64
#define M_   1024
#define RNN_ 1024
#define H_   512
#define D_   2048
#define NEG_INF_ (-1e9f)
#define MCHUNKS_ 8                 // chunks of M for the weighted-sum partials
#define MCH_     (M_ / MCHUNKS_)   // 128 rows per chunk

typedef float v2f __attribute__((ext_vector_type(2)));
typedef float v4f __attribute__((ext_vector_type(4)));
typedef float v8f __attribute__((ext_vector_type(8)));

// ---------------------------------------------------------------------------
// Fast device math
// ---------------------------------------------------------------------------
__device__ __forceinline__ float fast_exp(float x) {
  return __builtin_amdgcn_exp2f(x * 1.4426950408889634f);
}

__device__ __forceinline__ float fast_tanh(float x) {
#if __has_builtin(__builtin_amdgcn_tanhf)
  return __builtin_amdgcn_tanhf(x);        // v_tanh_f32 (gfx1250 TRANS op)
#else
  // tanh(x) = 1 - 2/(exp(2x)+1); overflow-safe (exp->inf => 1, exp->0 => -1)
  float e = __builtin_amdgcn_exp2f(x * 2.885390081777927f); // 2*log2(e)
  return 1.0f - 2.0f / (e + 1.0f);
#endif
}

// ---------------------------------------------------------------------------
// Kernel 1: wah[B,H] = h[B,RNN] @ W_ah[H,RNN]^T   via V_WMMA_F32_16X16X4_F32
// One wave per 16x16 output tile. Grid = (B/16)*(H/16) = 4*32 = 128 blocks.
//
// ISA lane layouts (CDNA5 ISA):
//   A 16x4 f32 : lanes 0-15 -> M=lane, VGPR{0,1}=K{0,1}; lanes 16-31 -> K{2,3}
//   B 4x16 f32 : lanes 0-15 -> N=lane, VGPR{0,1}=K{0,1}; lanes 16-31 -> K{2,3}
//   C/D 16x16  : VGPR r -> M = r + (lane>=16 ? 8 : 0), N = lane&15
// ---------------------------------------------------------------------------
__global__ void __launch_bounds__(32)
wah_gemm_wmma(const float* __restrict__ h, const float* __restrict__ W_ah,
              float* __restrict__ wah) {
  const int tile = blockIdx.x;
  const int b0 = (tile >> 5) * 16;     // 32 H-tiles per B-tile row
  const int h0 = (tile & 31) * 16;

  const int lane = threadIdx.x;        // 0..31, full wave, no divergence
  const int idx  = lane & 15;          // A row / B col for this lane
  const int kk   = (lane >> 4) * 2;    // K-pair selector per half-wave

  const float* __restrict__ arow = h    + (size_t)(b0 + idx) * RNN_;
  const float* __restrict__ brow = W_ah + (size_t)(h0 + idx) * RNN_;

  v8f c = {0.f, 0.f, 0.f, 0.f, 0.f, 0.f, 0.f, 0.f};
  for (int k = 0; k < RNN_; k += 4) {
    v2f a, b;
    a.x = arow[k + kk];
    a.y = arow[k + kk + 1];
    b.x = brow[k + kk];
    b.y = brow[k + kk + 1];
    c = __builtin_amdgcn_wmma_f32_16x16x4_f32(
        /*neg_a=*/false, a, /*neg_b=*/false, b,
        /*c_mod=*/(short)0, c, /*reuse_a=*/false, /*reuse_b=*/false);
  }

  const int mbase = b0 + (lane >> 4) * 8;
#pragma unroll
  for (int r = 0; r < 8; ++r)
    wah[(size_t)(mbase + r) * H_ + h0 + idx] = c[r];
}

// ---------------------------------------------------------------------------
// Kernel 2: logits[b,m] = sum_h tanh(wah[b,h] + p[b,m,h]) * w_alpha[h], masked.
// Streams p_att_feats (128 MB) once with NT hints. 8 waves/block, 1 m per wave.
// Grid = B * M/8 = 8192 blocks of 256.
// ---------------------------------------------------------------------------
__global__ void __launch_bounds__(256)
logits_kernel(const float* __restrict__ wah, const float* __restrict__ p_att,
              const float* __restrict__ w_alpha, const int* __restrict__ mask,
              float* __restrict__ logits) {
  __shared__ __align__(16) float s_wah[H_];
  __shared__ __align__(16) float s_wal[H_];

  const int b   = blockIdx.x >> 7;       // 128 blocks per batch
  const int mg  = blockIdx.x & 127;
  const int tid = threadIdx.x;

  s_wah[tid]       = wah[(size_t)b * H_ + tid];
  s_wah[tid + 256] = wah[(size_t)b * H_ + tid + 256];
  s_wal[tid]       = w_alpha[tid];
  s_wal[tid + 256] = w_alpha[tid + 256];
  __syncthreads();

  const int lane = tid & 31;
  const int m    = mg * 8 + (tid >> 5);

  const v4f* __restrict__ prow =
      (const v4f*)(p_att + ((size_t)b * M_ + m) * H_);
  const v4f* s_wah4 = (const v4f*)s_wah;
  const v4f* s_wal4 = (const v4f*)s_wal;

  float sum = 0.f;
#pragma unroll
  for (int j = 0; j < H_ / (32 * 4); ++j) {   // 4 iterations of float4
    const int i4 = lane + j * 32;
    v4f p = __builtin_nontemporal_load(&prow[i4]);   // streamed once -> NT
    v4f w = s_wah4[i4];
    v4f a = s_wal4[i4];
#pragma unroll
    for (int c = 0; c < 4; ++c)
      sum += fast_tanh(p[c] + w[c]) * a[c];
  }

  // wave32 reduction
#pragma unroll
  for (int off = 16; off > 0; off >>= 1)
    sum += __shfl_xor(sum, off, 32);

  if (lane == 0) {
    const size_t o = (size_t)b * M_ + m;
    logits[o] = (mask[o] == 0) ? NEG_INF_ : sum;
  }
}

// ---------------------------------------------------------------------------
// Kernel 3: softmax over M per batch row. Grid = B blocks of 256 (4 m each).
// ---------------------------------------------------------------------------
__global__ void __launch_bounds__(256)
softmax_kernel(const float* __restrict__ logits, float* __restrict__ alpha) {
  __shared__ float red[8];
  const int b    = blockIdx.x;
  const int tid  = threadIdx.x;
  const int lane = tid & 31;
  const int wv   = tid >> 5;
  const float* __restrict__ row = logits + (size_t)b * M_;

  float v[4];
#pragma unroll
  for (int i = 0; i < 4; ++i) v[i] = row[tid + i * 256];

  float mx = fmaxf(fmaxf(v[0], v[1]), fmaxf(v[2], v[3]));
#pragma unroll
  for (int off = 16; off > 0; off >>= 1)
    mx = fmaxf(mx, __shfl_xor(mx, off, 32));
  if (lane == 0) red[wv] = mx;
  __syncthreads();
  float gmax = red[0];
#pragma unroll
  for (int w = 1; w < 8; ++w) gmax = fmaxf(gmax, red[w]);
  __syncthreads();

  float e[4];
  float s = 0.f;
#pragma unroll
  for (int i = 0; i < 4; ++i) { e[i] = fast_exp(v[i] - gmax); s += e[i]; }
#pragma unroll
  for (int off = 16; off > 0; off >>= 1)
    s += __shfl_xor(s, off, 32);
  if (lane == 0) red[wv] = s;
  __syncthreads();
  float gsum = 0.f;
#pragma unroll
  for (int w = 0; w < 8; ++w) gsum += red[w];

  const float inv = 1.0f / gsum;
  float* __restrict__ out = alpha + (size_t)b * M_;
#pragma unroll
  for (int i = 0; i < 4; ++i) out[tid + i * 256] = e[i] * inv;
}

// ---------------------------------------------------------------------------
// Kernel 4: partial weighted sums over M chunks (deterministic, no atomics).
// part[c][b][d] = sum_{m in chunk c} alpha[b,m] * feats[b,m,d]
// Streams att_feats (512 MB) once with NT hints.
// Grid = (B, MCHUNKS) blocks of 512; each thread owns one float4 of D.
// ---------------------------------------------------------------------------
__global__ void __launch_bounds__(512)
weighted_sum_partial(const float* __restrict__ alpha,
                     const float* __restrict__ feats,
                     float* __restrict__ part) {
  __shared__ float s_a[MCH_];
  const int b   = blockIdx.x;
  const int c   = blockIdx.y;
  const int tid = threadIdx.x;

  if (tid < MCH_) s_a[tid] = alpha[(size_t)b * M_ + c * MCH_ + tid];
  __syncthreads();

  const v4f* __restrict__ frow =
      (const v4f*)(feats + ((size_t)b * M_ + (size_t)c * MCH_) * D_);
  v4f acc = {0.f, 0.f, 0.f, 0.f};
  for (int mm = 0; mm < MCH_; ++mm) {
    v4f f = __builtin_nontemporal_load(&frow[(size_t)mm * (D_ / 4) + tid]);
    acc += s_a[mm] * f;
  }
  v4f* __restrict__ out = (v4f*)(part + ((size_t)c * B_ + b) * D_);
  out[tid] = acc;
}

// ---------------------------------------------------------------------------
// Kernel 5: reduce MCHUNKS partials into d_out (fixed order => deterministic).
// Grid = B*D/256 = 512 blocks of 256.
// ---------------------------------------------------------------------------
__global__ void __launch_bounds__(256)
reduce_partials(const float* __restrict__ part, float* __restrict__ out) {
  const size_t i = (size_t)blockIdx.x * 256 + threadIdx.x;   // over B*D
  float s = 0.f;
#pragma unroll
  for (int c = 0; c < MCHUNKS_; ++c)
    s += part[(size_t)c * (B_ * D_) + i];
  out[i] = s;
}

// ---------------------------------------------------------------------------
// Host-side launcher
// ---------------------------------------------------------------------------
extern "C" void kernel_launch(void* const* d_in, const int* in_sizes, int n_in,
                              void* d_out, int out_size, void* d_ws, size_t ws_size,
                              hipStream_t stream) {
  const float* h        = (const float*)d_in[0];  // [B, RNN]
  const float* att_f    = (const float*)d_in[1];  // [B, M, D]
  const int*   att_mask = (const int*)  d_in[2];  // [B, M]
  const float* p_att    = (const float*)d_in[3];  // [B, M, H]
  const float* W_ah     = (const float*)d_in[4];  // [H, RNN]
  const float* w_alpha  = (const float*)d_in[5];  // [H]
  float*       out      = (float*)d_out;          // [B, D]

  // Workspace layout (floats)
  float* ws     = (float*)d_ws;
  float* wah    = ws;                              // B*H     = 32768
  float* logits = wah    + (size_t)B_ * H_;        // B*M     = 65536
  float* alpha  = logits + (size_t)B_ * M_;        // B*M     = 65536
  float* part   = alpha  + (size_t)B_ * M_;        // MCHUNKS*B*D = 1048576

  // 1) wah = h @ W_ah^T  (WMMA f32 16x16x4)
  wah_gemm_wmma<<<(B_ / 16) * (H_ / 16), 32, 0, stream>>>(h, W_ah, wah);

  // 2) additive-attention logits (streams p_att_feats, NT)
  logits_kernel<<<B_ * (M_ / 8), 256, 0, stream>>>(wah, p_att, w_alpha,
                                                   att_mask, logits);

  // 3) masked softmax over M
  softmax_kernel<<<B_, 256, 0, stream>>>(logits, alpha);

  // 4) chunked weighted sum of region features (streams att_feats, NT)
  weighted_sum_partial<<<dim3(B_, MCHUNKS_), 512, 0, stream>>>(alpha, att_f,
                                                               part);

  // 5) deterministic reduction of partials into d_out
  reduce_partials<<<(B_ * D_) / 256, 256, 0, stream>>>(part, out);
}